// NeuralCF_72559177499383
// MI455X (gfx1250) — compile-verified
//
#include <hip/hip_runtime.h>
#include <hip/hip_bf16.h>
#include <math.h>

#define EMB     128
#define BATCH   16384
#define NNODES  100000
#define NEDGES  640000

typedef float v2f __attribute__((ext_vector_type(2)));
typedef float v8f __attribute__((ext_vector_type(8)));

__device__ __forceinline__ float wave_sum(float v) {
#pragma unroll
    for (int m = 16; m >= 1; m >>= 1) v += __shfl_xor(v, m, 32);
    return v;
}

// ---------------------------------------------------------------------------
// f32 WMMA GEMM: Y[M, 16*NT] = act(X[M,K] @ W[K, 16*NT] + bias), K % 64 == 0.
// One wave computes a 16-row tile across all NT column tiles.
// W is staged per-block into LDS as K-pairs {W[k,c],W[k+1,c]} (float2), row
// stride N+16 float2 (== 32 words mod 64 banks) so the b64 LDS reads of the
// two lane-halves land on disjoint bank halves (conflict-free).
// Inner loop: A is prefetched one k-step ahead (global b64), all NT B-frags
// are loaded into registers first (claused ds_load_2addr_b64, single
// s_wait_dscnt), then NT WMMAs issue back-to-back.
// A-frag (16x4 f32): lanes 0-15 hold row M=l, K={k,k+1}; lanes 16-31 K={k+2,k+3}
// B-frag (4x16 f32): lanes 0-15 hold col N=l, K={k,k+1}; lanes 16-31 K={k+2,k+3}
// C/D  (16x16 f32): VGPR v -> row (v + 8*half), col = l.
// ---------------------------------------------------------------------------
template <int NT>
__global__ void __launch_bounds__(256)
wmma_gemm_kernel(const float* __restrict__ X,
                 const float* __restrict__ W,
                 const float* __restrict__ bias,
                 float* __restrict__ Y,
                 int M, int K, int act) {
    constexpr int N   = NT * 16;
    constexpr int KC  = 64;         // K chunk staged in LDS
    constexpr int LDW = N + 16;     // float2 row stride (bank-tuned)
    __shared__ float2 wt2[(KC / 2) * LDW];

    const int tid   = threadIdx.x;
    const int wave  = tid >> 5;
    const int lane  = tid & 31;
    const int half  = lane >> 4;    // 0: lanes 0-15, 1: lanes 16-31
    const int l     = lane & 15;
    const int rowBase = (blockIdx.x * (blockDim.x >> 5) + wave) * 16;

    v8f acc[NT] = {};

    const int  row   = rowBase + l;
    const bool rowOK = (row < M);
    const int  rr    = rowOK ? row : (M - 1);   // clamp, mask after load
    const float* xrow = X + (size_t)rr * K;

    for (int kc = 0; kc < K; kc += KC) {
        // ---- cooperative transposed staging of W[kc:kc+KC, :] ----
        for (int i = tid; i < (KC / 2) * N; i += blockDim.x) {
            const int p = i / N;        // K-pair index
            const int c = i % N;        // column (coalesced)
            float2 w2;
            w2.x = W[(size_t)(kc + 2 * p)     * N + c];
            w2.y = W[(size_t)(kc + 2 * p + 1) * N + c];
            wt2[p * LDW + c] = w2;
        }
        __syncthreads();

        // ---- 16 k-steps of NT WMMAs from LDS, A prefetched 1 step ahead ----
        float2 a_nxt = *(const float2*)(xrow + kc + 2 * half);
#pragma unroll
        for (int k = 0; k < KC; k += 4) {
            v2f a;
            a.x = rowOK ? a_nxt.x : 0.0f;
            a.y = rowOK ? a_nxt.y : 0.0f;
            if (k + 4 < KC)
                a_nxt = *(const float2*)(xrow + kc + k + 4 + 2 * half);

            const float2* bp = wt2 + ((k >> 1) + half) * LDW + l;
            float2 breg[NT];
#pragma unroll
            for (int t = 0; t < NT; ++t) breg[t] = bp[t * 16];
#pragma unroll
            for (int t = 0; t < NT; ++t) {
                v2f b; b.x = breg[t].x; b.y = breg[t].y;
                acc[t] = __builtin_amdgcn_wmma_f32_16x16x4_f32(
                    false, a, false, b, (short)0, acc[t], false, false);
            }
        }
        __syncthreads();
    }

#pragma unroll
    for (int t = 0; t < NT; ++t) {
        const int   c  = t * 16 + l;
        const float bv = bias ? bias[c] : 0.0f;
#pragma unroll
        for (int v = 0; v < 8; ++v) {
            const int r = rowBase + v + 8 * half;
            if (r < M) {
                float o = acc[t][v] + bv;
                if (act) o = fmaxf(o, 0.0f);
                Y[(size_t)r * N + c] = o;
            }
        }
    }
}

// ---------------------------------------------------------------------------
// Weighted scatter-add of per-relation transformed source features.
// One wave per edge, float4 per lane (128 feats / 32 lanes).
// ---------------------------------------------------------------------------
__global__ void edge_scatter_kernel(const int* __restrict__ src,
                                    const int* __restrict__ dst,
                                    const int* __restrict__ etype,
                                    const float* __restrict__ eweight,
                                    const float* __restrict__ y0,
                                    const float* __restrict__ y1,
                                    float* __restrict__ out) {
    const int e = blockIdx.x * (blockDim.x >> 5) + (threadIdx.x >> 5);
    if (e >= NEDGES) return;
    const int lane = threadIdx.x & 31;
    const int s = src[e], d = dst[e];
    const float w = eweight[e];
    const float* y = (etype[e] == 0) ? y0 : y1;
    const float4 v = *(const float4*)(y + (size_t)s * EMB + lane * 4);
    float* o = out + (size_t)d * EMB + lane * 4;
    unsafeAtomicAdd(o + 0, w * v.x);
    unsafeAtomicAdd(o + 1, w * v.y);
    unsafeAtomicAdd(o + 2, w * v.z);
    unsafeAtomicAdd(o + 3, w * v.w);
}

// ---------------------------------------------------------------------------
// In-place ReLU + LayerNorm, one wave per 128-wide row.
// ---------------------------------------------------------------------------
__global__ void relu_ln_kernel(float* __restrict__ x,
                               const float* __restrict__ g,
                               const float* __restrict__ b,
                               int M) {
    const int row = blockIdx.x * (blockDim.x >> 5) + (threadIdx.x >> 5);
    if (row >= M) return;
    const int lane = threadIdx.x & 31;
    float* p = x + (size_t)row * EMB + lane * 4;
    float4 v = *(const float4*)p;
    v.x = fmaxf(v.x, 0.0f); v.y = fmaxf(v.y, 0.0f);
    v.z = fmaxf(v.z, 0.0f); v.w = fmaxf(v.w, 0.0f);
    const float mu = wave_sum(v.x + v.y + v.z + v.w) * (1.0f / EMB);
    const float dx = v.x - mu, dy = v.y - mu, dz = v.z - mu, dw = v.w - mu;
    const float var = wave_sum(dx * dx + dy * dy + dz * dz + dw * dw) * (1.0f / EMB);
    const float inv = rsqrtf(var + 1e-5f);
    const float4 gg = *(const float4*)(g + lane * 4);
    const float4 bb = *(const float4*)(b + lane * 4);
    float4 o;
    o.x = dx * inv * gg.x + bb.x;
    o.y = dy * inv * gg.y + bb.y;
    o.z = dz * inv * gg.z + bb.z;
    o.w = dw * inv * gg.w + bb.w;
    *(float4*)p = o;
}

// ---------------------------------------------------------------------------
// Gather u/v rows, L2-normalize, GMF product, concat(u,v). One wave per item.
// ---------------------------------------------------------------------------
__global__ void gather_gmf_kernel(const float* __restrict__ h2,
                                  const int* __restrict__ ui,
                                  const int* __restrict__ vi,
                                  float* __restrict__ gmf,
                                  float* __restrict__ hcat) {
    const int bidx = blockIdx.x * (blockDim.x >> 5) + (threadIdx.x >> 5);
    if (bidx >= BATCH) return;
    const int lane = threadIdx.x & 31;
    const int u = ui[bidx], w = vi[bidx];
    const float4 uv = *(const float4*)(h2 + (size_t)u * EMB + lane * 4);
    const float4 vv = *(const float4*)(h2 + (size_t)w * EMB + lane * 4);
    const float su = wave_sum(uv.x * uv.x + uv.y * uv.y + uv.z * uv.z + uv.w * uv.w);
    const float sv = wave_sum(vv.x * vv.x + vv.y * vv.y + vv.z * vv.z + vv.w * vv.w);
    const float rn = 1.0f / (fmaxf(sqrtf(su), 1e-12f) * fmaxf(sqrtf(sv), 1e-12f));
    float4 gg;
    gg.x = uv.x * vv.x * rn; gg.y = uv.y * vv.y * rn;
    gg.z = uv.z * vv.z * rn; gg.w = uv.w * vv.w * rn;
    *(float4*)(gmf + (size_t)bidx * EMB + lane * 4) = gg;
    *(float4*)(hcat + (size_t)bidx * 2 * EMB + lane * 4) = uv;
    *(float4*)(hcat + (size_t)bidx * 2 * EMB + EMB + lane * 4) = vv;
}

// ---------------------------------------------------------------------------
// Final sigmoid(concat(gmf, h3) @ out_w + out_b). One wave per item.
// ---------------------------------------------------------------------------
__global__ void score_kernel(const float* __restrict__ gmf,
                             const float* __restrict__ h3,
                             const float* __restrict__ out_w,
                             const float* __restrict__ out_b,
                             float* __restrict__ out) {
    const int bidx = blockIdx.x * (blockDim.x >> 5) + (threadIdx.x >> 5);
    if (bidx >= BATCH) return;
    const int lane = threadIdx.x & 31;
    const float4 gg = *(const float4*)(gmf + (size_t)bidx * EMB + lane * 4);
    const float4 ww = *(const float4*)(out_w + lane * 4);
    float acc = gg.x * ww.x + gg.y * ww.y + gg.z * ww.z + gg.w * ww.w;
    acc += h3[(size_t)bidx * 32 + lane] * out_w[EMB + lane];
    acc = wave_sum(acc) + out_b[0];
    if (lane == 0) out[bidx] = 1.0f / (1.0f + expf(-acc));
}

extern "C" void kernel_launch(void* const* d_in, const int* in_sizes, int n_in,
                              void* d_out, int out_size, void* d_ws, size_t ws_size,
                              hipStream_t stream) {
    const int*   user_idx = (const int*)d_in[0];
    const int*   item_idx = (const int*)d_in[1];
    const int*   eidx     = (const int*)d_in[2];   // [2, E]: row 0 = src, row 1 = dst
    const int*   etype    = (const int*)d_in[3];
    const float* eweight  = (const float*)d_in[4];
    const float* emb      = (const float*)d_in[5];
    const float* w1_rel   = (const float*)d_in[6];
    const float* w1_root  = (const float*)d_in[7];
    const float* b1       = (const float*)d_in[8];
    const float* ln_g     = (const float*)d_in[9];
    const float* ln_b     = (const float*)d_in[10];
    const float* w2_rel   = (const float*)d_in[11];
    const float* w2_root  = (const float*)d_in[12];
    const float* b2       = (const float*)d_in[13];
    const float* mlp_w0   = (const float*)d_in[14];
    const float* mlp_b0   = (const float*)d_in[15];
    const float* mlp_w1   = (const float*)d_in[16];
    const float* mlp_b1   = (const float*)d_in[17];
    const float* mlp_w2   = (const float*)d_in[18];
    const float* mlp_b2   = (const float*)d_in[19];
    const float* out_w    = (const float*)d_in[20];
    const float* out_b    = (const float*)d_in[21];
    float* out = (float*)d_out;

    char* ws = (char*)d_ws;
    const size_t NB = (size_t)NNODES * EMB * sizeof(float);   // 51.2 MB
    float* y0 = (float*)(ws);
    float* y1 = (float*)(ws + NB);
    float* h1 = (float*)(ws + 2 * NB);
    float* h2 = (float*)(ws + 3 * NB);
    // Scoring-phase buffers alias y0's region (free after layer-2 scatter).
    float* gmf  = (float*)(ws);
    float* hcat = (float*)(ws + (size_t)BATCH * EMB * 4);
    float* m0   = (float*)(ws + (size_t)BATCH * EMB * 4 + (size_t)BATCH * 2 * EMB * 4);
    float* m1   = (float*)((char*)m0 + (size_t)BATCH * 128 * 4);
    float* m2   = (float*)((char*)m1 + (size_t)BATCH * 64 * 4);

    const int nodeGrid  = (NNODES + 127) / 128;   // 8 waves x 16 rows per block
    const int batchGrid = (BATCH + 127) / 128;
    const int edgeGrid  = (NEDGES + 7) / 8;
    const int rowGridN  = (NNODES + 7) / 8;
    const int rowGridB  = (BATCH + 7) / 8;

    // ---- Layer 1 ----
    wmma_gemm_kernel<8><<<nodeGrid, 256, 0, stream>>>(emb, w1_rel,             nullptr, y0, NNODES, EMB, 0);
    wmma_gemm_kernel<8><<<nodeGrid, 256, 0, stream>>>(emb, w1_rel + EMB * EMB, nullptr, y1, NNODES, EMB, 0);
    wmma_gemm_kernel<8><<<nodeGrid, 256, 0, stream>>>(emb, w1_root,            b1,      h1, NNODES, EMB, 0);
    edge_scatter_kernel<<<edgeGrid, 256, 0, stream>>>(eidx, eidx + NEDGES, etype, eweight, y0, y1, h1);
    relu_ln_kernel<<<rowGridN, 256, 0, stream>>>(h1, ln_g, ln_b, NNODES);

    // ---- Layer 2 ----
    wmma_gemm_kernel<8><<<nodeGrid, 256, 0, stream>>>(h1, w2_rel,             nullptr, y0, NNODES, EMB, 0);
    wmma_gemm_kernel<8><<<nodeGrid, 256, 0, stream>>>(h1, w2_rel + EMB * EMB, nullptr, y1, NNODES, EMB, 0);
    wmma_gemm_kernel<8><<<nodeGrid, 256, 0, stream>>>(h1, w2_root,            b2,      h2, NNODES, EMB, 0);
    edge_scatter_kernel<<<edgeGrid, 256, 0, stream>>>(eidx, eidx + NEDGES, etype, eweight, y0, y1, h2);

    // ---- Scoring head ----
    gather_gmf_kernel<<<rowGridB, 256, 0, stream>>>(h2, user_idx, item_idx, gmf, hcat);
    wmma_gemm_kernel<8><<<batchGrid, 256, 0, stream>>>(hcat, mlp_w0, mlp_b0, m0, BATCH, 2 * EMB, 1);
    wmma_gemm_kernel<4><<<batchGrid, 256, 0, stream>>>(m0,   mlp_w1, mlp_b1, m1, BATCH, 128,     1);
    wmma_gemm_kernel<2><<<batchGrid, 256, 0, stream>>>(m1,   mlp_w2, mlp_b2, m2, BATCH, 64,      1);
    score_kernel<<<rowGridB, 256, 0, stream>>>(gmf, m2, out_w, out_b, out);
}